// CenterLoss_65609920413924
// MI455X (gfx1250) — compile-verified
//
#include <hip/hip_runtime.h>

// ---------------------------------------------------------------------------
// CenterLoss on MI455X (gfx1250, wave32)
//   logits: (N=4, C=19, H=512, W=1024) f32, target unused by reference
//   out:    scalar f32
// Phase 1: streaming pass (b128 loads, per-block LDS f32 atomics, global
//          f32 atomics into 76x3 accumulators)  -- bandwidth bound, ~159MB
// Phase 2: one-wave finalize; 76-norm reduction via V_WMMA_F32_16X16X4_F32
// ---------------------------------------------------------------------------

#define NCLASS 19
#define NSAMP  4
#define HW     (512 * 1024)      // 524288 pixels per sample
#define BLOCK  256
#define PPT    2                 // float4 pixel-groups per thread per block
#define QUADS_PER_BLOCK (BLOCK * PPT)          // 512 float4 groups
#define BLOCKS_PER_SAMPLE (HW / (QUADS_PER_BLOCK * 4))  // 256
#define NSEG   (NSAMP * NCLASS)  // 76 segments
#define ACCSZ  (NSEG * 3)        // 228 floats of workspace

typedef __attribute__((ext_vector_type(2))) float v2f;
typedef __attribute__((ext_vector_type(8))) float v8f;

__global__ void centerloss_zero(float* __restrict__ acc) {
    int i = threadIdx.x;
    if (i < ACCSZ) acc[i] = 0.0f;
}

__global__ __launch_bounds__(BLOCK) void centerloss_main(
        const float* __restrict__ logits, float* __restrict__ acc) {
    __shared__ float sacc[NCLASS * 3];
    const int tid = threadIdx.x;
    for (int i = tid; i < NCLASS * 3; i += BLOCK) sacc[i] = 0.0f;
    __syncthreads();

    const int n   = blockIdx.x / BLOCKS_PER_SAMPLE;
    const int blk = blockIdx.x % BLOCKS_PER_SAMPLE;
    const float* base = logits + (size_t)n * (size_t)NCLASS * (size_t)HW;

    for (int it = 0; it < PPT; ++it) {
        const size_t q   = (size_t)blk * QUADS_PER_BLOCK + (size_t)it * BLOCK + tid;
        const size_t off = q * 4;   // first pixel of this float4 group

        float s1[4], s2[4], best[4];
        int   bc[4];
        {
            const float4 v = *reinterpret_cast<const float4*>(base + off);
            const float vv[4] = {v.x, v.y, v.z, v.w};
#pragma unroll
            for (int j = 0; j < 4; ++j) {
                s1[j] = vv[j];
                s2[j] = vv[j] * vv[j];
                best[j] = vv[j];
                bc[j] = 0;
            }
        }
#pragma unroll
        for (int c = 1; c < NCLASS; ++c) {
            const float* p = base + (size_t)c * (size_t)HW + off;
            // plane stride is 2MB: software-prefetch the next block-iteration's
            // line on this plane (speculative; OOB prefetch is silently dropped)
            __builtin_prefetch((const void*)(p + 4 * BLOCK), 0, 0);
            const float4 u = *reinterpret_cast<const float4*>(p);
            const float uu[4] = {u.x, u.y, u.z, u.w};
#pragma unroll
            for (int j = 0; j < 4; ++j) {
                s1[j] += uu[j];
                s2[j]  = fmaf(uu[j], uu[j], s2[j]);
                if (uu[j] > best[j]) { best[j] = uu[j]; bc[j] = c; }  // first-max wins
            }
        }
#pragma unroll
        for (int j = 0; j < 4; ++j) {
            atomicAdd(&sacc[bc[j] * 3 + 0], s1[j]);   // ds_add_f32
            atomicAdd(&sacc[bc[j] * 3 + 1], s2[j]);
            atomicAdd(&sacc[bc[j] * 3 + 2], 1.0f);
        }
    }
    __syncthreads();
    for (int i = tid; i < NCLASS * 3; i += BLOCK)
        atomicAdd(&acc[n * NCLASS * 3 + i], sacc[i]); // global_atomic_add_f32
}

// One wave (32 lanes, EXEC all ones -> WMMA legal). Computes per-segment norms
// and reduces the 76 values (zero-padded to 128) with two accumulating
// f32 16x16x4 WMMAs against an all-ones B matrix: D[M][N] = sum_K A[M][K].
__global__ __launch_bounds__(32) void centerloss_finalize(
        const float* __restrict__ acc, float* __restrict__ out) {
    __shared__ float norm[128];
    __shared__ float part[32];
    const int tid = threadIdx.x;

    for (int i = tid; i < 128; i += 32) {
        float v = 0.0f;
        if (i < NSEG) {
            const float S1  = acc[i * 3 + 0];
            const float S2  = acc[i * 3 + 1];
            const float cnt = acc[i * 3 + 2];
            const float K   = fmaxf(cnt, 1.0f) * (float)NCLASS;
            const float sq  = fmaxf(S2 - S1 * S1 / K, 0.0f);
            v = (cnt > 0.0f) ? sqrtf(sq) : 0.0f;
        }
        norm[i] = v;
    }
    __syncthreads();

    // A-matrix 16x4 f32 layout (ISA 7.12.2): lanes 0-15 hold K=0,1 for M=lane;
    // lanes 16-31 hold K=2,3 for M=lane-16.  A[M][K] := vals[M*4+K].
    const int M  = tid & 15;
    const int kb = (tid < 16) ? 0 : 2;
    v2f a0, a1, b;
    a0.x = norm[M * 4 + kb];
    a0.y = norm[M * 4 + kb + 1];
    a1.x = norm[64 + M * 4 + kb];
    a1.y = norm[64 + M * 4 + kb + 1];
    b.x = 1.0f;  b.y = 1.0f;   // all-ones B: layout-independent

    v8f c = {0.0f, 0.0f, 0.0f, 0.0f, 0.0f, 0.0f, 0.0f, 0.0f};
    c = __builtin_amdgcn_wmma_f32_16x16x4_f32(false, a0, false, b,
                                              (short)0, c, false, false);
    c = __builtin_amdgcn_wmma_f32_16x16x4_f32(false, a1, false, b,
                                              (short)0, c, false, false);

    // Column N=0 of D: lane 0 vgpr r -> M=r (r=0..7); lane 16 vgpr r -> M=8+r.
    float r = c[0] + c[1] + c[2] + c[3] + c[4] + c[5] + c[6] + c[7];
    part[tid] = r;
    __syncthreads();
    if (tid == 0) {
        // sum_c mean_n norm = (1/N) * sum_{n,c} norm
        out[0] = (part[0] + part[16]) * (1.0f / (float)NSAMP);
    }
}

extern "C" void kernel_launch(void* const* d_in, const int* in_sizes, int n_in,
                              void* d_out, int out_size, void* d_ws, size_t ws_size,
                              hipStream_t stream) {
    const float* logits = (const float*)d_in[0];
    // d_in[1] (target) is unused by the reference computation.
    float* out = (float*)d_out;
    float* acc = (float*)d_ws;   // 228 floats

    centerloss_zero<<<1, 256, 0, stream>>>(acc);
    centerloss_main<<<NSAMP * BLOCKS_PER_SAMPLE, BLOCK, 0, stream>>>(logits, acc);
    centerloss_finalize<<<1, 32, 0, stream>>>(acc, out);
}